// RelationGAT_71167608094724
// MI455X (gfx1250) — compile-verified
//
#include <hip/hip_runtime.h>
#include <hip/hip_bf16.h>
#include <math.h>

typedef __attribute__((ext_vector_type(2))) float v2f;
typedef __attribute__((ext_vector_type(8))) float v8f;

#define B_DIM 1024
#define L_DIM 128
#define DM    256

// ---------------------------------------------------------------------------
// Kernel 1: relsum[b,:] = sum_l emb[items[b,l], :]
// 1024 blocks x 256 threads; items row cached in LDS; 1KB row reads are fully
// coalesced; software prefetch 8 rows ahead (global_prefetch_b8) to hide the
// gather latency.
// ---------------------------------------------------------------------------
__global__ void gather_sum_kernel(const float* __restrict__ emb,
                                  const int*   __restrict__ items,
                                  float*       __restrict__ relsum) {
  __shared__ int s_items[L_DIM];
  const int b   = blockIdx.x;
  const int tid = threadIdx.x;                 // 0..255 = embedding channel
  if (tid < L_DIM) s_items[tid] = items[b * L_DIM + tid];
  __syncthreads();

  constexpr int PF = 8;
  float acc = 0.0f;
#pragma unroll 4
  for (int l = 0; l < L_DIM; ++l) {
    if (l + PF < L_DIM)
      __builtin_prefetch(emb + (size_t)s_items[l + PF] * DM + tid, 0, 1);
    acc += emb[(size_t)s_items[l] * DM + tid];
  }
  relsum[(size_t)b * DM + tid] = acc;
}

// ---------------------------------------------------------------------------
// Kernel 2/3: C[M,N] = A[M,K] @ B[K,N], row-major fp32, V_WMMA_F32_16X16X4_F32.
// Compile-time dims -> pure shift/imm addressing, no tail code.
//
// One wave32 owns a 32(M) x 16(N) strip: two accumulators share each B
// fragment (halves the strided B loads; 2 independent WMMA chains hide the
// matrix-pipe latency).
//
// Fragment layouts (CDNA5 ISA 7.12.2):
//   A 16x4 : lanes 0-15 M=lane, VGPR0=K0 VGPR1=K1; lanes 16-31 K2/K3
//            -> contiguous float2 at A[row, k + 2*half] (one b64 load).
//   B 4x16 : rows striped across lanes (mirror of A along K).
//   C 16x16: VGPR r -> M = 8*half + r, N = lane&15.
// ---------------------------------------------------------------------------
template <int M, int N, int K>
__global__ void gemm_f32_wmma(const float* __restrict__ A,
                              const float* __restrict__ Bm,
                              float*       __restrict__ C) {
  const int lane  = threadIdx.x & 31;
  const int wave  = threadIdx.x >> 5;                // 0..7
  const int tileN = blockIdx.x * 16;
  const int tileM = (blockIdx.y * 8 + wave) * 32;    // 32-row strip per wave
  const int half  = lane >> 4;                       // 0: lanes 0-15, 1: 16-31
  const int l16   = lane & 15;

  const float* Arow0 = A + (size_t)(tileM + l16) * K + 2 * half;
  const float* Arow1 = Arow0 + (size_t)16 * K;
  const float* Bcol  = Bm + (size_t)(2 * half) * N + tileN + l16;

  v8f acc0 = {};
  v8f acc1 = {};
#pragma unroll 8
  for (int k = 0; k < K; k += 4) {
    v2f a0 = *(const v2f*)(Arow0 + k);               // global_load_b64
    v2f a1 = *(const v2f*)(Arow1 + k);               // global_load_b64
    v2f b;
    b.x = Bcol[(size_t)k * N];                       // coalesced b32
    b.y = Bcol[(size_t)(k + 1) * N];                 // coalesced b32
    // 8 args: (neg_a, A, neg_b, B, c_mod, C, reuse_a, reuse_b)
    acc0 = __builtin_amdgcn_wmma_f32_16x16x4_f32(
        false, a0, false, b, (short)0, acc0, false, false);
    acc1 = __builtin_amdgcn_wmma_f32_16x16x4_f32(
        false, a1, false, b, (short)0, acc1, false, false);
  }

  float* C0 = C + (size_t)(tileM + 8 * half) * N + tileN + l16;
  float* C1 = C0 + (size_t)16 * N;
#pragma unroll
  for (int r = 0; r < 8; ++r) {
    C0[(size_t)r * N] = acc0[r];
    C1[(size_t)r * N] = acc1[r];
  }
}

// ---------------------------------------------------------------------------
// Kernel 4: out[b,:] = selu(R[b,:]) / ||selu(R[b,:])||_2
// 1024 blocks x 256 threads, LDS tree reduction for the squared norm.
// ---------------------------------------------------------------------------
__global__ void selu_norm_kernel(const float* __restrict__ R,
                                 float*       __restrict__ out) {
  __shared__ float red[DM];
  const int b   = blockIdx.x;
  const int tid = threadIdx.x;

  const float kScale = 1.0507009873554805f;
  const float kAlpha = 1.6732632423543772f;

  float x = R[(size_t)b * DM + tid];
  float y = (x > 0.0f) ? kScale * x : kScale * kAlpha * (expf(x) - 1.0f);

  red[tid] = y * y;
  __syncthreads();
#pragma unroll
  for (int s = DM / 2; s > 0; s >>= 1) {
    if (tid < s) red[tid] += red[tid + s];
    __syncthreads();
  }
  const float inv = 1.0f / sqrtf(red[0]);
  out[(size_t)b * DM + tid] = y * inv;
}

// ---------------------------------------------------------------------------
// Host-side orchestration. Input order (setup_inputs):
//  0 item_embedding, 1 A, 2 D, 3 target_embedding, 4 atten_w0, 5 atten_w1,
//  6 atten_w2, 7 atten_bias, 8 alpha_w_w, 9 alpha_w_b, 10 items.
// Attention/entmax branch is provably a no-op (scores constant over L ->
// uniform weights -> c == rel), so only emb/A/D/items feed the output.
// D@(A@x) == (D@A)@x, avoiding the 1024^3 GEMM.
// ---------------------------------------------------------------------------
extern "C" void kernel_launch(void* const* d_in, const int* in_sizes, int n_in,
                              void* d_out, int out_size, void* d_ws, size_t ws_size,
                              hipStream_t stream) {
  const float* emb   = (const float*)d_in[0];
  const float* A     = (const float*)d_in[1];
  const float* D     = (const float*)d_in[2];
  const int*   items = (const int*)d_in[10];
  float*       out   = (float*)d_out;

  float* relsum = (float*)d_ws;                 // [1024, 256]
  float* t      = relsum + (size_t)B_DIM * DM;  // [1024, 256]
  float* r      = t + (size_t)B_DIM * DM;       // [1024, 256]

  // 1) rel0 = sum_L emb[items]
  gather_sum_kernel<<<dim3(B_DIM), dim3(DM), 0, stream>>>(emb, items, relsum);

  // 2) t = A @ rel0     grid: 16 N-tiles x (M / (32*8 waves)) = (16, 4)
  gemm_f32_wmma<B_DIM, DM, B_DIM>
      <<<dim3(DM / 16, B_DIM / 256), dim3(256), 0, stream>>>(A, relsum, t);

  // 3) r = D @ t
  gemm_f32_wmma<B_DIM, DM, B_DIM>
      <<<dim3(DM / 16, B_DIM / 256), dim3(256), 0, stream>>>(D, t, r);

  // 4) out = selu(r) row-normalized
  selu_norm_kernel<<<dim3(B_DIM), dim3(DM), 0, stream>>>(r, out);
}